// SubbandFilter_5102421148009
// MI455X (gfx1250) — compile-verified
//
#include <hip/hip_runtime.h>
#include <hip/hip_bf16.h>
#include <math.h>

// ---------------- problem constants (fixed by the reference) ----------------
#define LEN     442368      // samples per (b,c) channel
#define LD      27648       // decimated length (LEN/16)
#define KB      16          // number of bands == WMMA M dimension
#define NB      48000       // band-filter taps
#define NU      12000       // up-filter taps
#define FACTOR  16
#define C1      23999       // (NB-1)//2  analysis centering offset
#define CH      192         // stage-1 K-chunk (48000 % 192 == 0)
#define CJ      96          // stage-2 K-chunk (768 = 8*96)
#define PI_F    3.14159265358979f
#define INV_SR  (1.0 / 44100.0)

typedef __attribute__((ext_vector_type(2))) float v2f;
typedef __attribute__((ext_vector_type(8))) float v8f;

#if defined(__has_builtin)
#if __has_builtin(__builtin_amdgcn_wmma_f32_16x16x4_f32)
#define HAVE_WMMA_F32 1
#endif
#endif

__device__ __forceinline__ v8f wmma_f32(v2f a, v2f b, v8f c) {
#ifdef HAVE_WMMA_F32
  return __builtin_amdgcn_wmma_f32_16x16x4_f32(false, a, false, b, (short)0, c,
                                               false, false);
#else
  for (int i = 0; i < 8; ++i) c[i] = fmaf(a.x, b.x, fmaf(a.y, b.y, c[i]));
  return c;
#endif
}

// 2*pi * frac(f * tnorm): f64 multiply + fract only -- no f64 division chain.
// Arguments are non-negative, so frac via floor is exact enough here.
__device__ __forceinline__ float wrap_angle(float f, double tnorm) {
  double z = (double)f * tnorm;
  z -= floor(z);
  return 6.283185307179586f * (float)z;
}

// ---------------------------------------------------------------------------
// Stage 0: wi[k,t] = (H{w_k})[t] via antisymmetric windowed Hilbert FIR.
// ---------------------------------------------------------------------------
__global__ __launch_bounds__(256) void k_hilbert(const float* __restrict__ w,
                                                 float* __restrict__ wi) {
  int idx = blockIdx.x * 256 + threadIdx.x;
  if (idx >= KB * NB) return;
  int k = idx / NB, t = idx % NB;
  const float* wk = w + k * NB;
  float acc = 0.f;
  for (int n = 1; n <= 511; n += 2) {
    float win = 0.54f + 0.46f * __cosf(PI_F * (float)n / 512.0f);
    float hn = (2.0f / (PI_F * (float)n)) * win;
    float a = (t - n >= 0) ? wk[t - n] : 0.f;
    float b = (t + n < NB) ? wk[t + n] : 0.f;
    acc = fmaf(hn, a - b, acc);
  }
  wi[idx] = acc;
}

// ---------------------------------------------------------------------------
// Stage 1: analysis GEMM at decimated rate + demodulation.
//   yd[k,n] = sum_t (wr + j*wi)[k,t] * x[16n + C1 - t],  then * e^{-j a(k,n)}
// 8 waves/block, 2 output tiles per wave (A fragments shared across B tiles).
// x window stored REVERSED in LDS so B pairs are ascending, aligned b64 loads.
// ---------------------------------------------------------------------------
__global__ __launch_bounds__(256) void k_analysis(
    const float* __restrict__ x, const float* __restrict__ wr,
    const float* __restrict__ wi, const float* __restrict__ fc,
    float* __restrict__ ydr, float* __restrict__ ydi) {
  __shared__ float sWr[KB][CH];
  __shared__ float sWi[KB][CH];
  __shared__ float sX[4288];                 // 16*255 + CH = 4272 used, reversed

  const int tid = threadIdx.x;
  const int bc = blockIdx.y;                 // 0..3
  const int Nbase = blockIdx.x * 256;        // block covers 256 decimated n
  const int wv = tid >> 5;
  const int ln = tid & 31;
  const int nl = ln & 15;
  const int hl = ln >> 4;
  const int roff0 = 16 * (255 - (32 * wv + nl));   // tile 0 (reversed base)
  const float* xc = x + bc * LEN;

  v8f accR[2] = {};
  v8f accI[2] = {};

  for (int t0 = 0; t0 < NB; t0 += CH) {
    __syncthreads();
    if (t0 + CH < NB) {                      // advisory prefetch of next chunk
      const int pr = (tid >> 4) * NB + (t0 + CH) + (tid & 15) * 12;
      __builtin_prefetch(&wr[pr], 0, 0);
      __builtin_prefetch(&wi[pr], 0, 0);
    }
    for (int i = tid; i < KB * CH; i += 256) {
      int k = i / CH, tl = i % CH;
      sWr[k][tl] = wr[k * NB + t0 + tl];
      sWi[k][tl] = wi[k * NB + t0 + tl];
    }
    const int xmax = 16 * (Nbase + 255) + C1 - t0;   // reversed: sX[i]=x[xmax-i]
    for (int i = tid; i < 4272; i += 256) {
      int gi = xmax - i;
      sX[i] = (gi >= 0 && gi < LEN) ? xc[gi] : 0.f;
    }
    __syncthreads();

#pragma unroll 4
    for (int tt = 0; tt < CH; tt += 4) {
      const int ta = tt + 2 * hl;            // even -> aligned b64 pairs
      v2f ar, ai, b0, b1;
      ar.x = sWr[nl][ta];        ar.y = sWr[nl][ta + 1];
      ai.x = sWi[nl][ta];        ai.y = sWi[nl][ta + 1];
      b0.x = sX[roff0 + ta];     b0.y = sX[roff0 + ta + 1];
      b1.x = sX[roff0 - 256 + ta]; b1.y = sX[roff0 - 256 + ta + 1];
      accR[0] = wmma_f32(ar, b0, accR[0]);
      accI[0] = wmma_f32(ai, b0, accI[0]);
      accR[1] = wmma_f32(ar, b1, accR[1]);
      accI[1] = wmma_f32(ai, b1, accI[1]);
    }
  }

  // demodulate y *= exp(-j*2*pi*f_k*(16n/sr)); store planar complex
#pragma unroll
  for (int s = 0; s < 2; ++s) {
    const int n = Nbase + 32 * wv + 16 * s + nl;
    const double tnorm = (double)(16 * n) * INV_SR;  // hoisted time factor
#pragma unroll
    for (int v = 0; v < 8; ++v) {
      int k = v + 8 * hl;                    // C/D layout: M = v + 8*(lane/16)
      float ang = wrap_angle(fc[k], tnorm);
      float cs, sn;
      __sincosf(ang, &sn, &cs);
      float yr = accR[s][v], yi = accI[s][v];
      int g = (bc * KB + k) * LD + n;
      ydr[g] = fmaf(yr, cs, yi * sn);
      ydi[g] = fmaf(yi, cs, -yr * sn);
    }
  }
}

// ---------------------------------------------------------------------------
// Stage 2: polyphase synthesis GEMM + remodulation + band reduction.
//   s[16q+p] = 16 * sum_d up[16d+p+5999] * yd[k, q-d],  d in [-374,375]
// 8 waves/block, 2 q-tiles per wave (B Toeplitz shared across the A tiles).
// up-phase row stored REVERSED with zero guards -> branchless inner loop.
// ---------------------------------------------------------------------------
__global__ __launch_bounds__(256) void k_synthesis(
    const float* __restrict__ ydr, const float* __restrict__ ydi,
    const float* __restrict__ up, const float* __restrict__ fc,
    float* __restrict__ out) {
  __shared__ float sUp[800];                 // buf[16+r] = upp[749-r], guards=0
  __shared__ float sYr[KB][336];
  __shared__ float sYi[KB][336];

  const int tid = threadIdx.x;
  const int p = blockIdx.y;                  // phase 0..15
  const int bc = blockIdx.z;                 // 0..3
  const int q0blk = blockIdx.x * 256;
  const int wv = tid >> 5;
  const int ln = tid & 31;
  const int nl = ln & 15;
  const int hl = ln >> 4;

  for (int i = tid; i < 800; i += 256) {     // reversed up-phase row + guards
    float v = 0.f;
    int r = i - 16;
    if (r >= 0 && r < 750) {
      int gi = 16 * (749 - r) + p + 15;      // upp[d+374] = up[16d+p+5999]
      v = (gi < NU) ? up[gi] : 0.f;
    }
    sUp[i] = v;
  }

  v8f accR[2] = {};
  v8f accI[2] = {};
  for (int j0 = 0; j0 < 768; j0 += CJ) {
    __syncthreads();
    const int qwin = q0blk - 375 + j0;       // window qoff in [0,336)
    if (j0 + CJ < 768) {                     // advisory prefetch of next window
      int qn = qwin + CJ + (tid & 15) * 21;
      qn = (qn < 0) ? 0 : ((qn >= LD) ? LD - 1 : qn);
      const int pr = (bc * KB + (tid >> 4)) * LD + qn;
      __builtin_prefetch(&ydr[pr], 0, 0);
      __builtin_prefetch(&ydi[pr], 0, 0);
    }
    for (int i = tid; i < KB * 336; i += 256) {
      int k = i / 336, qo = i % 336;
      int q = qwin + qo;
      bool ok = (q >= 0 && q < LD);
      int g = (bc * KB + k) * LD + q;
      sYr[k][qo] = ok ? ydr[g] : 0.f;
      sYi[k][qo] = ok ? ydi[g] : 0.f;
    }
    __syncthreads();

#pragma unroll 4
    for (int jj = 0; jj < CJ; jj += 4) {
      const int ja = jj + 2 * hl;
      const int q0 = 32 * wv + ja;           // qoff of tile 0 A fragment
      const int ub = 16 + j0 + ja - nl;      // reversed Toeplitz index
      v2f ar0, ai0, ar1, ai1, b;
      ar0.x = sYr[nl][q0];       ar0.y = sYr[nl][q0 + 1];
      ai0.x = sYi[nl][q0];       ai0.y = sYi[nl][q0 + 1];
      ar1.x = sYr[nl][q0 + 16];  ar1.y = sYr[nl][q0 + 17];
      ai1.x = sYi[nl][q0 + 16];  ai1.y = sYi[nl][q0 + 17];
      b.x = sUp[ub];             b.y = sUp[ub + 1];
      accR[0] = wmma_f32(ar0, b, accR[0]);
      accI[0] = wmma_f32(ai0, b, accI[0]);
      accR[1] = wmma_f32(ar1, b, accR[1]);
      accI[1] = wmma_f32(ai1, b, accI[1]);
    }
  }

  // remodulate, take real, reduce the 16 bands held in the tile, store
#pragma unroll
  for (int s = 0; s < 2; ++s) {
    const int m = 16 * (q0blk + 32 * wv + 16 * s + nl) + p;
    const double tnorm = (double)m * INV_SR;           // hoisted time factor
    float part = 0.f;
#pragma unroll
    for (int v = 0; v < 8; ++v) {
      int k = v + 8 * hl;
      float ang = wrap_angle(fc[k], tnorm);
      float cs, sn;
      __sincosf(ang, &sn, &cs);
      part = fmaf(accR[s][v], cs, part);
      part = fmaf(-accI[s][v], sn, part);
    }
    part *= (float)FACTOR;
    part += __shfl_xor(part, 16, 32);        // add bands 8..15 lane-half
    if (hl == 0) out[bc * LEN + m] = part;
  }
}

// ---------------------------------------------------------------------------
extern "C" void kernel_launch(void* const* d_in, const int* in_sizes, int n_in,
                              void* d_out, int out_size, void* d_ws,
                              size_t ws_size, hipStream_t stream) {
  const float* x  = (const float*)d_in[0];   // (2,2,442368)
  const float* w  = (const float*)d_in[1];   // (16,48000)
  const float* fc = (const float*)d_in[2];   // (16,)
  const float* up = (const float*)d_in[3];   // (12000,)
  float* out = (float*)d_out;

  // workspace: wi (3.07MB) | ydr (7.08MB) | ydi (7.08MB)
  float* wi  = (float*)d_ws;
  float* ydr = (float*)((char*)d_ws + (size_t)KB * NB * 4);
  float* ydi = (float*)((char*)d_ws + (size_t)KB * NB * 4 +
                        (size_t)4 * KB * LD * 4);

  k_hilbert<<<(KB * NB + 255) / 256, 256, 0, stream>>>(w, wi);
  k_analysis<<<dim3(LD / 256, 4), 256, 0, stream>>>(x, w, wi, fc, ydr, ydi);
  k_synthesis<<<dim3(LD / 256, 16, 4), 256, 0, stream>>>(ydr, ydi, up, fc, out);
}